// BatchPPRFeatures_65352222376166
// MI455X (gfx1250) — compile-verified
//
#include <hip/hip_runtime.h>
#include <stdint.h>

#define D_DIM 128
#define K_NBR 32
#define KCH   8                       // neighbors gathered per pipeline stage
#define NSTG  (K_NBR / KCH)           // 4 stages
#define WPB   8                       // waves per block (wave32)
#define TPB   (WPB * 32)
#define ROW_BYTES   (D_DIM * 4)       // 512 B per feature row
#define STAGE_BYTES (KCH * ROW_BYTES) // 4 KB per stage buffer

typedef float v4f __attribute__((ext_vector_type(4)));

__device__ __forceinline__ float bcast_f(float v, int lane) {
    return __int_as_float(__builtin_amdgcn_readlane(__float_as_int(v), lane));
}

// Issue one pipeline stage: 8 gathered feature rows -> LDS buffer `buf`.
// Each global_load_async_to_lds_b128 moves one full 512B row (32 lanes x 16B),
// per-lane source offset = row*512 + lane*16, saddr = base of x.
__device__ __forceinline__ void issue_stage(const float* x, uint32_t ldsBase,
                                            uint32_t laneB, int idxVec,
                                            int s, int buf) {
    const uint32_t dst = ldsBase + (uint32_t)buf * STAGE_BYTES;
#pragma unroll
    for (int j = 0; j < KCH; ++j) {
        const int      row  = __builtin_amdgcn_readlane(idxVec, s * KCH + j);
        const uint32_t voff = (uint32_t)row * ROW_BYTES + laneB;
        const uint32_t la   = dst + (uint32_t)j * ROW_BYTES + laneB;
        asm volatile("global_load_async_to_lds_b128 %0, %1, %2"
                     :: "v"(la), "v"(voff), "s"(x) : "memory");
    }
}

__global__ __launch_bounds__(TPB) void BatchPPRFeatures_kernel(
    const float* __restrict__ x,
    const int*   __restrict__ ppr_idx,
    const float* __restrict__ ppr_scores,
    float*       __restrict__ out,
    int n_nodes, int n_waves)
{
    // Per-wave double-buffered staging tiles: [wave][buf][8 rows x 128 floats]
    __shared__ float tile[WPB][2][KCH * D_DIM];

    const int lane = threadIdx.x & 31;
    const int wv   = threadIdx.x >> 5;
    const int gw   = blockIdx.x * WPB + wv;

    // Low 32 bits of a flat shared-aperture address are the LDS byte offset.
    const uint32_t ldsBase = (uint32_t)(uintptr_t)&tile[wv][0][0];
    const uint32_t laneB   = (uint32_t)lane * 16u;

    int n = gw;
    if (n >= n_nodes) return;

    // Lane k owns neighbor k's index and score (coalesced 128B, streaming -> NT).
    int   myIdx = __builtin_nontemporal_load(&ppr_idx  [(size_t)n * K_NBR + lane]);
    float mySc  = __builtin_nontemporal_load(&ppr_scores[(size_t)n * K_NBR + lane]);

    // Prime the pipe: stage 0 of the first node into buffer 0.
    issue_stage(x, ldsBase, laneB, myIdx, 0, 0);

    while (true) {
        const int  n_next   = n + n_waves;
        const bool has_next = n_next < n_nodes;

        // Fetch next node's metadata early (overlaps with this node's compute).
        int   nxIdx = 0;
        float nxSc  = 0.f;
        if (has_next) {
            nxIdx = __builtin_nontemporal_load(&ppr_idx  [(size_t)n_next * K_NBR + lane]);
            nxSc  = __builtin_nontemporal_load(&ppr_scores[(size_t)n_next * K_NBR + lane]);
        }

        float ax = 0.f, ay = 0.f, az = 0.f, aw = 0.f;

        // Stages 0..2: issue stage s+1, wait until only it remains in flight
        // (in-order async completion => stage s is fully in LDS), then reduce.
#pragma unroll
        for (int s = 0; s < NSTG - 1; ++s) {
            issue_stage(x, ldsBase, laneB, myIdx, s + 1, (s + 1) & 1);
            asm volatile("s_wait_asynccnt 0x8" ::: "memory");

            const float* buf = &tile[wv][s & 1][0];
#pragma unroll
            for (int j = 0; j < KCH; ++j) {
                const float w = bcast_f(mySc, s * KCH + j);
                const v4f   v = ((const v4f*)(buf + j * D_DIM))[lane]; // ds_load_b128
                ax = fmaf(w, v.x, ax);
                ay = fmaf(w, v.y, ay);
                az = fmaf(w, v.z, az);
                aw = fmaf(w, v.w, aw);
            }
        }

        // Stage 3: keep the pipe full across nodes — buf0 was last read in
        // stage 2 (program order), so the next node's stage 0 can issue now.
        if (has_next) {
            issue_stage(x, ldsBase, laneB, nxIdx, 0, 0);
            asm volatile("s_wait_asynccnt 0x8" ::: "memory");
        } else {
            asm volatile("s_wait_asynccnt 0x0" ::: "memory");
        }
        {
            const int    s   = NSTG - 1;
            const float* buf = &tile[wv][s & 1][0];
#pragma unroll
            for (int j = 0; j < KCH; ++j) {
                const float w = bcast_f(mySc, s * KCH + j);
                const v4f   v = ((const v4f*)(buf + j * D_DIM))[lane];
                ax = fmaf(w, v.x, ax);
                ay = fmaf(w, v.y, ay);
                az = fmaf(w, v.z, az);
                aw = fmaf(w, v.w, aw);
            }
        }

        // Streaming 512B row store (16B/lane), NT so it never evicts x from L2.
        v4f o;
        o.x = ax; o.y = ay; o.z = az; o.w = aw;
        __builtin_nontemporal_store(o, (v4f*)(out + (size_t)n * D_DIM) + lane);

        if (!has_next) break;
        n     = n_next;
        myIdx = nxIdx;
        mySc  = nxSc;
    }
}

extern "C" void kernel_launch(void* const* d_in, const int* in_sizes, int n_in,
                              void* d_out, int out_size, void* d_ws, size_t ws_size,
                              hipStream_t stream) {
    const float* x   = (const float*)d_in[0];
    const int*   idx = (const int*)  d_in[1];
    const float* sc  = (const float*)d_in[2];
    float*       out = (float*)d_out;

    const int n_nodes = in_sizes[0] / D_DIM;   // N = 100000
    const int blocks  = 1600;                  // 12800 waves => ~8 nodes per wave
    const int n_waves = blocks * WPB;

    hipLaunchKernelGGL(BatchPPRFeatures_kernel, dim3(blocks), dim3(TPB), 0, stream,
                       x, idx, sc, out, n_nodes, n_waves);
}